// ConvTemporalGraphical_11501922418891
// MI455X (gfx1250) — compile-verified
//
#include <hip/hip_runtime.h>
#include <hip/hip_bf16.h>
#include <stdint.h>

typedef float v2f __attribute__((ext_vector_type(2)));
typedef float v8f __attribute__((ext_vector_type(8)));
typedef unsigned int u32x4 __attribute__((ext_vector_type(4)));
typedef int i32x4 __attribute__((ext_vector_type(4)));
typedef int i32x8 __attribute__((ext_vector_type(8)));

#define EPS_ 0.001f

// Problem sizes (fixed by the reference)
constexpr int N_  = 16;
constexpr int CIN = 64;
constexpr int CO  = 64;
constexpr int T_  = 64;
constexpr int V_  = 384;
constexpr int COLS = T_ * V_;                     // 24576
constexpr long long PLANE = (long long)V_ * V_;   // 147456

// -------------------------------------------------------------------------
// TDM: DMA a 2D tile (tile_d1 rows x tile_d0 elements, f32) from global into
// LDS at lds_off, row stride in the tensor = stride0 elements.
// Descriptor layout per CDNA5 ISA (D# groups 0/1).
// -------------------------------------------------------------------------
__device__ __forceinline__ void tdm_load_2d_f32(uint32_t lds_off,
                                                const float* gptr,
                                                uint32_t tensor_d0,
                                                uint32_t tensor_d1,
                                                uint32_t tile_d0,
                                                uint32_t tile_d1,
                                                uint64_t stride0) {
    const uint64_t ga = (uint64_t)gptr;
    u32x4 g0;
    g0[0] = 1u;                                         // count=1 (valid), no gather
    g0[1] = lds_off;                                    // lds_addr (bytes)
    g0[2] = (uint32_t)ga;                               // global_addr[31:0]
    g0[3] = (uint32_t)((ga >> 32) & 0x01FFFFFFu)        // global_addr[56:32]
          | (2u << 30);                                 // type=2 ("image")

    i32x8 g1;
    g1[0] = (int)(2u << 16);                            // wg_mask=0, data_size=2 (4B)
    g1[1] = (int)((tensor_d0 & 0xFFFFu) << 16);         // tensor_dim0[15:0] @bits63:48
    g1[2] = (int)(((tensor_d0 >> 16) & 0xFFFFu)         // tensor_dim0[31:16]
          |  ((tensor_d1 & 0xFFFFu) << 16));            // tensor_dim1[15:0]
    g1[3] = (int)(((tensor_d1 >> 16) & 0xFFFFu)         // tensor_dim1[31:16]
          |  ((tile_d0 & 0xFFFFu) << 16));              // tile_dim0
    g1[4] = (int)(tile_d1 & 0xFFFFu);                   // tile_dim1; tile_dim2=0
    g1[5] = (int)(uint32_t)(stride0 & 0xFFFFFFFFu);     // tensor_dim0_stride[31:0]
    g1[6] = (int)(uint32_t)((stride0 >> 32) & 0xFFFFu); // stride[47:32]; dim1_stride lo=0
    g1[7] = 0;                                          // tensor_dim1_stride hi

    i32x4 z4 = {0, 0, 0, 0};
#if __clang_major__ >= 23
    i32x8 z8 = {0, 0, 0, 0, 0, 0, 0, 0};
    __builtin_amdgcn_tensor_load_to_lds(g0, g1, z4, z4, z8, 0);
#else
    __builtin_amdgcn_tensor_load_to_lds(g0, g1, z4, z4, 0);
#endif
}

// -------------------------------------------------------------------------
// Kernel 1: x' = conv_w @ x + conv_b   (per-batch GEMM M=64, K=64, N=24576)
// B panel (64 x 16 f32, row stride COLS) staged to LDS via TDM, then
// V_WMMA_F32_16X16X4_F32 with K stepped by 4. One 16x16 tile per wave.
// -------------------------------------------------------------------------
__global__ void k_conv1x1_x(const float* __restrict__ x,
                            const float* __restrict__ W,
                            const float* __restrict__ bias,
                            float* __restrict__ xp) {
    __shared__ float sB[CIN * 16];                  // 4 KB: [c][16 cols]

    const int lane = threadIdx.x & 31;
    const int wave = threadIdx.x >> 5;              // 0..3 -> M tile (out channels)
    const int colT = blockIdx.x;                    // 0..1535
    const int n    = blockIdx.y;                    // batch
    const int col0 = colT * 16;
    const int m0   = wave * 16;
    const int half = lane >> 4;                     // 0 | 1
    const int l    = lane & 15;

    const float* xb = x + (long long)n * CIN * COLS;

    if (wave == 0) {
        tdm_load_2d_f32((uint32_t)(uintptr_t)&sB[0], xb + col0,
                        /*tensor_d0=*/COLS, /*tensor_d1=*/CIN,
                        /*tile_d0=*/16, /*tile_d1=*/CIN,
                        /*stride0=*/(uint64_t)COLS);
        __builtin_amdgcn_s_wait_tensorcnt((short)0);
    }
    __syncthreads();

    v8f acc = {};
#pragma unroll 4
    for (int k0 = 0; k0 < CIN; k0 += 4) {
        const int ka = k0 + half * 2;
        v2f a, b;
        a.x = W[(m0 + l) * CIN + ka];
        a.y = W[(m0 + l) * CIN + ka + 1];
        b.x = sB[ka * 16 + l];
        b.y = sB[(ka + 1) * 16 + l];
        acc = __builtin_amdgcn_wmma_f32_16x16x4_f32(
            false, a, false, b, (short)0, acc, false, false);
    }

    float* xpo = xp + (long long)n * CO * COLS;
#pragma unroll
    for (int r = 0; r < 8; ++r) {
        const int o = m0 + r + half * 8;
        xpo[(long long)o * COLS + col0 + l] = acc[r] + bias[o];
    }
}

// -------------------------------------------------------------------------
// Kernel 2: 3-layer channel MLP (7->16->32->64) + ReLU + mask, writes Am
// (unnormalized An) directly into the An slot of d_out. Weights in LDS.
// -------------------------------------------------------------------------
__global__ void k_mlp_mask(const float* __restrict__ A,
                           const float* __restrict__ w1, const float* __restrict__ b1,
                           const float* __restrict__ w2, const float* __restrict__ b2,
                           const float* __restrict__ w3, const float* __restrict__ b3,
                           float* __restrict__ Am) {
    __shared__ float sw1[16 * 7], sb1[16];
    __shared__ float sw2[32 * 16], sb2[32];
    __shared__ float sw3[64 * 32], sb3[64];

    for (int i = threadIdx.x; i < 16 * 7;  i += blockDim.x) sw1[i] = w1[i];
    for (int i = threadIdx.x; i < 16;      i += blockDim.x) sb1[i] = b1[i];
    for (int i = threadIdx.x; i < 32 * 16; i += blockDim.x) sw2[i] = w2[i];
    for (int i = threadIdx.x; i < 32;      i += blockDim.x) sb2[i] = b2[i];
    for (int i = threadIdx.x; i < 64 * 32; i += blockDim.x) sw3[i] = w3[i];
    for (int i = threadIdx.x; i < 64;      i += blockDim.x) sb3[i] = b3[i];
    __syncthreads();

    const long long idx = (long long)blockIdx.x * blockDim.x + threadIdx.x;
    const long long total = (long long)N_ * PLANE;
    if (idx >= total) return;

    const int  w = (int)(idx % V_);
    const int  v = (int)((idx / V_) % V_);
    const int  n = (int)(idx / PLANE);

    const float* Ab = A + ((long long)n * 8 * V_ + v) * V_ + w;

    float h0[7];
#pragma unroll
    for (int c = 0; c < 7; ++c) h0[c] = Ab[(long long)c * PLANE];
    const float mask = Ab[7LL * PLANE];

    float l1[16];
#pragma unroll
    for (int o = 0; o < 16; ++o) {
        float s = sb1[o];
#pragma unroll
        for (int c = 0; c < 7; ++c) s = fmaf(sw1[o * 7 + c], h0[c], s);
        l1[o] = fmaxf(s, 0.f);
    }

    float l2[32];
#pragma unroll
    for (int o = 0; o < 32; ++o) {
        float s = sb2[o];
#pragma unroll
        for (int c = 0; c < 16; ++c) s = fmaf(sw2[o * 16 + c], l1[c], s);
        l2[o] = fmaxf(s, 0.f);
    }

    float* Ap = Am + ((long long)n * CO * V_ + v) * V_ + w;
#pragma unroll
    for (int o = 0; o < 64; ++o) {
        float s = sb3[o];
#pragma unroll
        for (int c = 0; c < 32; ++c) s = fmaf(sw3[o * 32 + c], l2[c], s);
        Ap[(long long)o * PLANE] = fmaxf(s, 0.f) * mask;
    }
}

// -------------------------------------------------------------------------
// Kernel 3: Dl[n,c,w] = 1 / (sum_v Am[n,c,v,w] + eps)
// -------------------------------------------------------------------------
__global__ void k_colsum(const float* __restrict__ Am, float* __restrict__ Dl) {
    const long long idx = (long long)blockIdx.x * blockDim.x + threadIdx.x;
    const long long total = (long long)N_ * CO * V_;   // 393216
    if (idx >= total) return;
    const int w = (int)(idx % V_);
    const long long nc = idx / V_;
    const float* p = Am + nc * PLANE + w;
    float s = 0.f;
    for (int v = 0; v < V_; ++v) s += p[(long long)v * V_];
    Dl[idx] = 1.f / (s + EPS_);
}

// -------------------------------------------------------------------------
// Kernel 4: in-place normalize An = Am * Dl (float4 along w)
// -------------------------------------------------------------------------
__global__ void k_norm(float* __restrict__ An, const float* __restrict__ Dl) {
    const long long idx = (long long)blockIdx.x * blockDim.x + threadIdx.x;
    const long long total4 = (long long)N_ * CO * PLANE / 4;
    if (idx >= total4) return;
    const long long e = idx * 4;
    const int w = (int)(e % V_);
    const long long nc = (e / V_) / V_;
    const float* dl = Dl + nc * V_ + w;
    float4 d = *(const float4*)(An + e);
    d.x *= dl[0]; d.y *= dl[1]; d.z *= dl[2]; d.w *= dl[3];
    *(float4*)(An + e) = d;
}

// -------------------------------------------------------------------------
// Kernel 5: out[n,c,t,w] = sum_v x'[n,c,t,v] * An[n,c,v,w]
// Per-(n,c) GEMM M=64(t) K=384(v) N=384(w). The 384x16 B panel of An is
// DMA'd once per block into LDS via TENSOR_LOAD_TO_LDS, then all 4 waves
// (covering the 4 M tiles) run the WMMA K-loop out of LDS.
// -------------------------------------------------------------------------
__global__ void k_gemm_out(const float* __restrict__ xp,
                           const float* __restrict__ An,
                           float* __restrict__ out) {
    __shared__ float sB[V_ * 16];                   // 24 KB: [v][16 cols]

    const int lane = threadIdx.x & 31;
    const int wave = threadIdx.x >> 5;              // t tile
    const int wt   = blockIdx.x;                    // w tile: 0..23
    const int c    = blockIdx.y;
    const int n    = blockIdx.z;
    const int half = lane >> 4;
    const int l    = lane & 15;
    const int t0   = wave * 16;
    const int w0   = wt * 16;

    const float* Ap = xp + (long long)(n * CO + c) * T_ * V_;   // [t][v]
    const float* Bp = An + (long long)(n * CO + c) * PLANE;     // [v][w]

    if (wave == 0) {
        tdm_load_2d_f32((uint32_t)(uintptr_t)&sB[0], Bp + w0,
                        /*tensor_d0=*/V_, /*tensor_d1=*/V_,
                        /*tile_d0=*/16, /*tile_d1=*/V_,
                        /*stride0=*/(uint64_t)V_);
        __builtin_amdgcn_s_wait_tensorcnt((short)0);
    }
    __syncthreads();

    v8f acc = {};
#pragma unroll 8
    for (int k0 = 0; k0 < V_; k0 += 4) {
        const int ka = k0 + half * 2;
        v2f a, b;
        a.x = Ap[(long long)(t0 + l) * V_ + ka];
        a.y = Ap[(long long)(t0 + l) * V_ + ka + 1];
        b.x = sB[ka * 16 + l];
        b.y = sB[(ka + 1) * 16 + l];
        // prefetch hint for the next K block of A
        __builtin_prefetch(&Ap[(long long)(t0 + l) * V_ + ka + 32], 0, 0);
        acc = __builtin_amdgcn_wmma_f32_16x16x4_f32(
            false, a, false, b, (short)0, acc, false, false);
    }

    float* op = out + (long long)(n * CO + c) * T_ * V_;
#pragma unroll
    for (int r = 0; r < 8; ++r)
        op[(long long)(t0 + r + half * 8) * V_ + w0 + l] = acc[r];
}

// -------------------------------------------------------------------------
extern "C" void kernel_launch(void* const* d_in, const int* in_sizes, int n_in,
                              void* d_out, int out_size, void* d_ws, size_t ws_size,
                              hipStream_t stream) {
    const float* x      = (const float*)d_in[0];
    const float* A      = (const float*)d_in[1];
    const float* conv_w = (const float*)d_in[2];
    const float* conv_b = (const float*)d_in[3];
    const float* w1     = (const float*)d_in[4];
    const float* b1     = (const float*)d_in[5];
    const float* w2     = (const float*)d_in[6];
    const float* b2     = (const float*)d_in[7];
    const float* w3     = (const float*)d_in[8];
    const float* b3     = (const float*)d_in[9];

    float* out = (float*)d_out;                               // (16,64,64,384)
    float* An  = out + (long long)N_ * CO * T_ * V_;          // (16,64,384,384)

    float* xp  = (float*)d_ws;                                // (16,64,64,384)
    float* Dl  = xp + (long long)N_ * CO * T_ * V_;           // (16,64,384)

    // 1) x' = conv_w @ x + b
    {
        dim3 grid(COLS / 16, N_);     // 1536 x 16
        k_conv1x1_x<<<grid, 128, 0, stream>>>(x, conv_w, conv_b, xp);
    }
    // 2) MLP + mask -> Am (stored in An slot)
    {
        const long long total = (long long)N_ * PLANE;        // 2359296
        const int blk = 256;
        const int grd = (int)((total + blk - 1) / blk);
        k_mlp_mask<<<grd, blk, 0, stream>>>(A, w1, b1, w2, b2, w3, b3, An);
    }
    // 3) column sums -> Dl
    {
        const long long total = (long long)N_ * CO * V_;
        const int blk = 256;
        k_colsum<<<(int)((total + blk - 1) / blk), blk, 0, stream>>>(An, Dl);
    }
    // 4) normalize in place
    {
        const long long total4 = (long long)N_ * CO * PLANE / 4;
        const int blk = 256;
        k_norm<<<(int)((total4 + blk - 1) / blk), blk, 0, stream>>>(An, Dl);
    }
    // 5) out = x' @ An
    {
        dim3 grid(V_ / 16, CO, N_);   // 24 x 64 x 16
        k_gemm_out<<<grid, 128, 0, stream>>>(xp, An, out);
    }
}